// BipartiteTransformerConvolution_40200893891316
// MI455X (gfx1250) — compile-verified
//
#include <hip/hip_runtime.h>
#include <hip/hip_bf16.h>
#include <math.h>

#define DMODEL 128

typedef float v2f __attribute__((ext_vector_type(2)));
typedef float v8f __attribute__((ext_vector_type(8)));

// ---------------- monotone float<->uint encoding for atomicMax on floats ----
__device__ __forceinline__ unsigned int fenc(float x) {
    unsigned int u = __float_as_uint(x);
    return (u & 0x80000000u) ? ~u : (u | 0x80000000u);
}
__device__ __forceinline__ float fdec(unsigned int u) {
    return (u & 0x80000000u) ? __uint_as_float(u & 0x7FFFFFFFu)
                             : __uint_as_float(~u);
}

// ---------------- WMMA fp32 GEMM: C[M,128] = A[M,K] @ W[K,128] (+bias, ReLU) ----
// REGB=true : wave keeps its 16-column W strip entirely in VGPRs (K/4 x v2f),
//             k-loop FULLY unrolled so breg[] indices are constants (no scratch
//             spill), one global_load_b64 per v_wmma in steady state.
// REGB=false: stream B fragments from cache each k-step (used for K=256 where
//             a register-resident strip would exceed comfortable VGPR budget).
// REQUIRES M % 16 == 0 (true for all shapes here: 50000, 500000).
template<int K, bool REGB, bool HAS_BIAS, bool RELU>
__global__ __launch_bounds__(256) void wmma_gemm(
    const float* __restrict__ A, const float* __restrict__ W,
    const float* __restrict__ bias, float* __restrict__ C, int Mtiles)
{
    const int lane = threadIdx.x & 31;
    const int half = lane >> 4;                 // 0/1: half-wave
    const int lr   = lane & 15;
    const int n0   = (threadIdx.x >> 5) << 4;   // wave id * 16 -> column strip

    v2f breg[REGB ? (K / 4) : 1];
    if constexpr (REGB) {
#pragma unroll
        for (int kk = 0; kk < K / 4; ++kk) {
            const float* wp = W + (size_t)(4 * kk + 2 * half) * DMODEL + n0 + lr;
            breg[kk].x = wp[0];
            breg[kk].y = wp[DMODEL];
        }
    }
    const float bv = HAS_BIAS ? bias[n0 + lr] : 0.0f;
    const float* Wp = W + (size_t)(2 * half) * DMODEL + n0 + lr;

    const int mstride = gridDim.x << 4;         // rows advanced per grid step

    for (int t = blockIdx.x; t < Mtiles; t += gridDim.x) {
        const int m0 = t << 4;
        const float* Ap = A + (size_t)(m0 + lr) * K;

        // hint the next tile's A rows into cache (global_prefetch_b8)
        __builtin_prefetch(Ap + (size_t)mstride * K, 0, 1);

        v8f acc = {0.f, 0.f, 0.f, 0.f, 0.f, 0.f, 0.f, 0.f};
        if constexpr (REGB) {
#pragma unroll
            for (int kk = 0; kk < K / 4; ++kk) {
                v2f a;  // A 16x4: lanes0-15 K={4kk,4kk+1}, lanes16-31 K={4kk+2,4kk+3}
                a.x = Ap[4 * kk + 2 * half];
                a.y = Ap[4 * kk + 2 * half + 1];
                acc = __builtin_amdgcn_wmma_f32_16x16x4_f32(
                    false, a, false, breg[kk], (short)0, acc, false, false);
            }
        } else {
#pragma unroll 8
            for (int kk = 0; kk < K / 4; ++kk) {
                v2f a, b;
                a.x = Ap[4 * kk + 2 * half];
                a.y = Ap[4 * kk + 2 * half + 1];
                b.x = Wp[(size_t)(4 * kk) * DMODEL];
                b.y = Wp[(size_t)(4 * kk + 1) * DMODEL];
                acc = __builtin_amdgcn_wmma_f32_16x16x4_f32(
                    false, a, false, b, (short)0, acc, false, false);
            }
        }

        float* Cp = C + (size_t)m0 * DMODEL + n0 + lr;
#pragma unroll
        for (int i = 0; i < 8; ++i) {           // VGPR i = row 8*half + i
            float v = acc[i] + bv;
            if (RELU) v = fmaxf(v, 0.0f);
            Cp[(size_t)((half << 3) + i) * DMODEL] = v;
        }
    }
}

// ---------------- init scratch (amax encoded -> 0 == below all encodings) ----
__global__ void init_kernel(unsigned int* amax, float* denom, float* outagg,
                            int nAmax, int nOut)
{
    int i = blockIdx.x * blockDim.x + threadIdx.x;
    if (i < nAmax) { amax[i] = 0u; denom[i] = 0.0f; }
    if (i < nOut)  outagg[i] = 0.0f;
}

// ---------------- per-edge attention logits + segment max ------------------
// one wave per edge; lane l owns elements [4l,4l+3] -> head = l/8
__global__ __launch_bounds__(256) void edge_logits_kernel(
    const int* __restrict__ src, const int* __restrict__ dst,
    const float* __restrict__ q, const float* __restrict__ kn,
    const float* __restrict__ ee, float* __restrict__ alpha,
    unsigned int* __restrict__ amax, int E)
{
    int e = blockIdx.x * 8 + (threadIdx.x >> 5);
    if (e >= E) return;
    int lane = threadIdx.x & 31;
    int s = src[e], d = dst[e];

    float4 qv = ((const float4*)(q  + (size_t)d * DMODEL))[lane];
    float4 kv = ((const float4*)(kn + (size_t)s * DMODEL))[lane];
    float4 ev = ((const float4*)(ee + (size_t)e * DMODEL))[lane];

    float ps = qv.x * (kv.x + ev.x) + qv.y * (kv.y + ev.y)
             + qv.z * (kv.z + ev.z) + qv.w * (kv.w + ev.w);
    // reduce within each 8-lane group (one head)
    ps += __shfl_xor(ps, 4, 32);
    ps += __shfl_xor(ps, 2, 32);
    ps += __shfl_xor(ps, 1, 32);

    if ((lane & 7) == 0) {
        int h = lane >> 3;
        float a = ps * 0.17677669529663687f;  // 1/sqrt(32)
        alpha[(size_t)e * 4 + h] = a;
        atomicMax(&amax[(size_t)d * 4 + h], fenc(a));
    }
}

// ---------------- per-edge exp + scatter-accumulate ------------------------
__global__ __launch_bounds__(256) void edge_accum_kernel(
    const int* __restrict__ src, const int* __restrict__ dst,
    const float* __restrict__ vn, const float* __restrict__ ee,
    const float* __restrict__ alpha, const unsigned int* __restrict__ amax,
    float* __restrict__ denom, float* __restrict__ outagg, int E)
{
    int e = blockIdx.x * 8 + (threadIdx.x >> 5);
    if (e >= E) return;
    int lane = threadIdx.x & 31;
    int s = src[e], d = dst[e];
    int h = lane >> 3;

    float a  = alpha[(size_t)e * 4 + h];
    float m  = fdec(amax[(size_t)d * 4 + h]);
    float ex = expf(a - m);
    if ((lane & 7) == 0) atomicAdd(&denom[(size_t)d * 4 + h], ex);

    float4 vv = ((const float4*)(vn + (size_t)s * DMODEL))[lane];
    float4 ev = ((const float4*)(ee + (size_t)e * DMODEL))[lane];
    float* op = outagg + (size_t)d * DMODEL + lane * 4;
    atomicAdd(op + 0, ex * (vv.x + ev.x));
    atomicAdd(op + 1, ex * (vv.y + ev.y));
    atomicAdd(op + 2, ex * (vv.z + ev.z));
    atomicAdd(op + 3, ex * (vv.w + ev.w));
}

// ---------------- per-node: softmax-normalize, beta gate, LayerNorm, concat -
__global__ __launch_bounds__(256) void node_finalize_kernel(
    const float* __restrict__ outagg, const float* __restrict__ denom,
    const float* __restrict__ xr, const float* __restrict__ right,
    const float* __restrict__ Wbeta, const float* __restrict__ ln_g,
    const float* __restrict__ ln_b, float* __restrict__ hcat, int Nr)
{
    int r = blockIdx.x * 8 + (threadIdx.x >> 5);
    if (r >= Nr) return;
    int lane = threadIdx.x & 31;
    int h = lane >> 3;

    float4 o = ((const float4*)(outagg + (size_t)r * DMODEL))[lane];
    float dn = denom[(size_t)r * 4 + h] + 1e-16f;
    o.x /= dn; o.y /= dn; o.z /= dn; o.w /= dn;

    float4 x  = ((const float4*)(xr + (size_t)r * DMODEL))[lane];
    float4 w0 = ((const float4*)(Wbeta))[lane];
    float4 w1 = ((const float4*)(Wbeta + DMODEL))[lane];
    float4 w2 = ((const float4*)(Wbeta + 2 * DMODEL))[lane];

    float bsum = o.x * w0.x + o.y * w0.y + o.z * w0.z + o.w * w0.w
               + x.x * w1.x + x.y * w1.y + x.z * w1.z + x.w * w1.w
               + (o.x - x.x) * w2.x + (o.y - x.y) * w2.y
               + (o.z - x.z) * w2.z + (o.w - x.w) * w2.w;
    for (int off = 16; off; off >>= 1) bsum += __shfl_xor(bsum, off, 32);
    float beta = 1.0f / (1.0f + expf(-bsum));

    float4 msg;
    msg.x = beta * x.x + (1.0f - beta) * o.x;
    msg.y = beta * x.y + (1.0f - beta) * o.y;
    msg.z = beta * x.z + (1.0f - beta) * o.z;
    msg.w = beta * x.w + (1.0f - beta) * o.w;

    float s = msg.x + msg.y + msg.z + msg.w;
    for (int off = 16; off; off >>= 1) s += __shfl_xor(s, off, 32);
    float mu = s * (1.0f / DMODEL);

    float v = (msg.x - mu) * (msg.x - mu) + (msg.y - mu) * (msg.y - mu)
            + (msg.z - mu) * (msg.z - mu) + (msg.w - mu) * (msg.w - mu);
    for (int off = 16; off; off >>= 1) v += __shfl_xor(v, off, 32);
    float rstd = rsqrtf(v * (1.0f / DMODEL) + 1e-5f);

    float4 g = ((const float4*)ln_g)[lane];
    float4 b = ((const float4*)ln_b)[lane];
    float4 nrm;
    nrm.x = (msg.x - mu) * rstd * g.x + b.x;
    nrm.y = (msg.y - mu) * rstd * g.y + b.y;
    nrm.z = (msg.z - mu) * rstd * g.z + b.z;
    nrm.w = (msg.w - mu) * rstd * g.w + b.w;

    float4 rf = ((const float4*)(right + (size_t)r * DMODEL))[lane];
    float* hb = hcat + (size_t)r * (2 * DMODEL);
    ((float4*)hb)[lane] = nrm;
    ((float4*)(hb + DMODEL))[lane] = rf;
}

static inline int grid_for(int Mtiles) {
    return Mtiles < 2048 ? Mtiles : 2048;   // amortize register-resident W strip
}

extern "C" void kernel_launch(void* const* d_in, const int* in_sizes, int n_in,
                              void* d_out, int out_size, void* d_ws, size_t ws_size,
                              hipStream_t stream)
{
    const float* left  = (const float*)d_in[0];
    const int*   eidx  = (const int*)  d_in[1];
    const float* efeat = (const float*)d_in[2];
    const float* right = (const float*)d_in[3];
    const float* Wq    = (const float*)d_in[4];
    const float* bq    = (const float*)d_in[5];
    const float* Wk    = (const float*)d_in[6];
    const float* bk    = (const float*)d_in[7];
    const float* Wv    = (const float*)d_in[8];
    const float* bv    = (const float*)d_in[9];
    const float* We    = (const float*)d_in[10];
    const float* Wskip = (const float*)d_in[11];
    const float* bskip = (const float*)d_in[12];
    const float* Wbeta = (const float*)d_in[13];
    const float* ln_g  = (const float*)d_in[14];
    const float* ln_b  = (const float*)d_in[15];
    const float* W1    = (const float*)d_in[16];
    const float* b1    = (const float*)d_in[17];
    const float* W2    = (const float*)d_in[18];
    const float* b2    = (const float*)d_in[19];

    const int NL = in_sizes[0] / DMODEL;
    const int E  = in_sizes[2] / DMODEL;
    const int NR = in_sizes[3] / DMODEL;
    const int* src = eidx;
    const int* dst = eidx + E;

    const int tilesNL = NL / 16, tilesNR = NR / 16, tilesE = E / 16;

    // ---- workspace carve-up (floats) ----
    float* p = (float*)d_ws;
    float* q_buf  = p;  p += (size_t)NR * DMODEL;
    float* kn     = p;  p += (size_t)NL * DMODEL;
    float* vn     = p;  p += (size_t)NL * DMODEL;
    float* xr     = p;  p += (size_t)NR * DMODEL;
    float* ee     = p;  p += (size_t)E  * DMODEL;
    float* alpha  = p;  p += (size_t)E  * 4;
    unsigned int* amax = (unsigned int*)p; p += (size_t)NR * 4;
    float* denom  = p;  p += (size_t)NR * 4;
    float* outagg = p;  p += (size_t)NR * DMODEL;
    float* hcat   = p;  p += (size_t)NR * 2 * DMODEL;
    float* t1     = p;  p += (size_t)NR * DMODEL;

    dim3 blk(256);

    // dense projections (fp32 WMMA GEMMs, register-resident W strips)
    wmma_gemm<DMODEL, true, true,  false><<<grid_for(tilesNR), blk, 0, stream>>>(right, Wq,    bq,    q_buf, tilesNR);
    wmma_gemm<DMODEL, true, true,  false><<<grid_for(tilesNL), blk, 0, stream>>>(left,  Wk,    bk,    kn,    tilesNL);
    wmma_gemm<DMODEL, true, true,  false><<<grid_for(tilesNL), blk, 0, stream>>>(left,  Wv,    bv,    vn,    tilesNL);
    wmma_gemm<DMODEL, true, true,  false><<<grid_for(tilesNR), blk, 0, stream>>>(right, Wskip, bskip, xr,    tilesNR);
    wmma_gemm<DMODEL, true, false, false><<<grid_for(tilesE ), blk, 0, stream>>>(efeat, We,    nullptr, ee,  tilesE);

    // zero/neg-inf scratch
    int nInit = NR * DMODEL;
    init_kernel<<<(nInit + 255) / 256, blk, 0, stream>>>(amax, denom, outagg, NR * 4, nInit);

    // sparse attention phase
    edge_logits_kernel<<<(E + 7) / 8, blk, 0, stream>>>(src, dst, q_buf, kn, ee, alpha, amax, E);
    edge_accum_kernel <<<(E + 7) / 8, blk, 0, stream>>>(src, dst, vn, ee, alpha, amax, denom, outagg, E);

    // beta gate + layernorm + concat
    node_finalize_kernel<<<(NR + 7) / 8, blk, 0, stream>>>(outagg, denom, xr, right,
                                                           Wbeta, ln_g, ln_b, hcat, NR);

    // output MLP (WMMA GEMMs; K=256 streams B to stay spill-free)
    wmma_gemm<2 * DMODEL, false, true, true ><<<grid_for(tilesNR), blk, 0, stream>>>(hcat, W1, b1, t1, tilesNR);
    wmma_gemm<DMODEL,     true,  true, false><<<grid_for(tilesNR), blk, 0, stream>>>(t1,   W2, b2, (float*)d_out, tilesNR);
}